// SpatialKNNEdge_37495064494461
// MI455X (gfx1250) — compile-verified
//
#include <hip/hip_runtime.h>

typedef __attribute__((ext_vector_type(2))) float v2f;
typedef __attribute__((ext_vector_type(4))) float v4f;
typedef __attribute__((ext_vector_type(8))) float v8f;

#define KNN_K 16
#define NFEAT 128
#define INF_F 1e30f

// ---------------------------------------------------------------------------
// Kernel 1: zero the 256MB adjacency with wide non-temporal stores (BW bound).
// ---------------------------------------------------------------------------
__global__ void zero_out_kernel(v4f* __restrict__ out, size_t n4) {
  size_t idx = (size_t)blockIdx.x * blockDim.x + threadIdx.x;
  const size_t stride = (size_t)gridDim.x * blockDim.x;
  const v4f z = {0.f, 0.f, 0.f, 0.f};
  for (; idx < n4; idx += stride)
    __builtin_nontemporal_store(z, &out[idx]);
}

// ---------------------------------------------------------------------------
// Kernel 2: KNN via V_WMMA_F32_16X16X4_F32 + register top-16 + sparse scatter.
// One wave (32 threads) per (batch, 16-sink tile).
//   A (16x4): lane m   = (sx, sy)     lane m+16 = (sz, |s|^2)
//   B (4x16): lane j   = (-2px,-2py)  lane j+16 = (-2pz, 1)
//   => C[row][col] = |s_row|^2 - 2*dot(s_row, p_col);  d2 = C + |p_col|^2
// ---------------------------------------------------------------------------
__global__ __launch_bounds__(32) void spatial_knn_kernel(
    const float* __restrict__ nodes, const int* __restrict__ T,
    const int* __restrict__ taus, float* __restrict__ adj, int N)
{
  const int b    = blockIdx.y;
  const int i0   = blockIdx.x << 4;         // first sink row of this tile
  const int lane = threadIdx.x;
  const int n    = lane & 15;
  const bool hiL = lane >= 16;

  const int tb  = T[b];
  const int tau = taus[b];
  if (i0 >= tau) return;                    // wave-uniform: tile fully invalid
  const int jmax = min(N, tb + tau);        // number of valid source rows

  const float* base = nodes + (size_t)b * N * NFEAT;

  // --- A matrix (sink positions for rows i0..i0+15), clip(T+i, 0, N-1)
  const int g = min(tb + i0 + n, N - 1);
  const float* sp = base + (size_t)g * NFEAT;
  const float sx = sp[0], sy = sp[1], sz = sp[2];
  const float s2 = sx * sx + sy * sy + sz * sz;
  v2f A;
  A.x = hiL ? sz : sx;
  A.y = hiL ? s2 : sy;

  // --- per-lane (lanes 0..15) running top-K for row i0+n, register resident
  float dk[KNN_K];
  int   ik[KNN_K];
#pragma unroll
  for (int t = 0; t < KNN_K; ++t) { dk[t] = INF_F; ik[t] = -1; }

  __shared__ float d2s[16 * 16];

  for (int j0 = 0; j0 < jmax; j0 += 16) {
    const int j  = j0 + n;
    const int jc = min(j, N - 1);
    const float* pp = base + (size_t)jc * NFEAT;
    // prefetch next chunk's candidate row (L2-resident most of the time)
    __builtin_prefetch(base + (size_t)min(j + 16, N - 1) * NFEAT, 0, 1);
    const float px = pp[0], py = pp[1], pz = pp[2];
    const float p2 = px * px + py * py + pz * pz;

    v2f Bm;
    Bm.x = hiL ? (-2.0f * pz) : (-2.0f * px);
    Bm.y = hiL ? 1.0f         : (-2.0f * py);

    v8f C = {};
    // D = A x B + 0 : emits v_wmma_f32_16x16x4_f32
    C = __builtin_amdgcn_wmma_f32_16x16x4_f32(
        /*neg_a=*/false, A, /*neg_b=*/false, Bm,
        /*c_mod=*/(short)0, C, /*reuse_a=*/false, /*reuse_b=*/false);

    const bool colValid = (j < jmax);
#pragma unroll
    for (int v = 0; v < 8; ++v) {
      const int row = v + (hiL ? 8 : 0);
      const float d2 = colValid ? (C[v] + p2) : INF_F;
      d2s[row * 16 + n] = d2;
    }
    __syncthreads();

    // --- selection: lanes 0..15 each own one sink row
    if (!hiL) {
      const float* rowp = &d2s[n * 16];
      for (int c = 0; c < 16; ++c) {
        const float d = rowp[c];
        if (d < dk[KNN_K - 1]) {            // rarely taken -> cheap scan
          dk[KNN_K - 1] = d;
          ik[KNN_K - 1] = j0 + c;
          // one bubble pass restores sorted order (list was sorted before)
#pragma unroll
          for (int t = KNN_K - 1; t > 0; --t) {
            if (dk[t] < dk[t - 1]) {
              const float td = dk[t]; dk[t] = dk[t - 1]; dk[t - 1] = td;
              const int   ti = ik[t]; ik[t] = ik[t - 1]; ik[t - 1] = ti;
            }
          }
        }
      }
    }
    __syncthreads();
  }

  // --- scatter: adj[b][i][j] = 1 for selected neighbors with j < i, i < tau
  if (!hiL) {
    const int i = i0 + n;
    if (i < tau) {
      float* rowOut = adj + ((size_t)b * N + i) * (size_t)N;
#pragma unroll
      for (int t = 0; t < KNN_K; ++t)
        if (ik[t] >= 0 && ik[t] < i) rowOut[ik[t]] = 1.0f;
    }
  }
}

// ---------------------------------------------------------------------------
extern "C" void kernel_launch(void* const* d_in, const int* in_sizes, int n_in,
                              void* d_out, int out_size, void* d_ws, size_t ws_size,
                              hipStream_t stream) {
  (void)n_in; (void)d_ws; (void)ws_size;
  const float* nodes = (const float*)d_in[0];
  const int*   T     = (const int*)d_in[1];
  const int*   taus  = (const int*)d_in[2];
  const int B = in_sizes[1];                       // len(T)
  const int N = in_sizes[0] / (B * NFEAT);         // 2048
  float* adj = (float*)d_out;

  // 1) zero-fill output (256 MB) at full HBM bandwidth
  const size_t n4 = (size_t)out_size / 4;
  zero_out_kernel<<<8192, 256, 0, stream>>>((v4f*)d_out, n4);

  // 2) WMMA-based KNN + sparse scatter
  dim3 grid(N / 16, B);
  spatial_knn_kernel<<<grid, 32, 0, stream>>>(nodes, T, taus, adj, N);
}